// NCELoss_14465449853062
// MI455X (gfx1250) — compile-verified
//
#include <hip/hip_runtime.h>
#include <math.h>

// Problem constants (from the reference)
#define B_DIM   256
#define L_DIM   50
#define E_DIM   256
#define V_DIM   100000
#define K_DIM   500
#define M_TOTAL (B_DIM * L_DIM)   // 12800 rows, divisible by 32
#define N_PAD   512               // 500 noise cols padded to 512
#define LOG_V   11.512925464970229f   // log(100000)
#define LOG_K   6.214608098422191f    // log(500)
#define INV_BL  (1.0f / 12800.0f)
#define BACKOFF 1e-10f

// Workspace layout (floats):
//  [0] sum(noise)   [1] sum(clip(noise/sum,1e-10))
//  [WS_NCOL .. +512)  per-noise-column constant: bias - logV - logprob_noise - logK
//  [WS_BMAT .. +512*256) gathered, zero-padded B panel (row-major by column, stride E)
#define WS_NCOL 8
#define WS_BMAT 1024

typedef __attribute__((ext_vector_type(2))) float v2f;
typedef __attribute__((ext_vector_type(8))) float v8f;
typedef int v4i __attribute__((vector_size(16)));   // matches builtin's param pointee

__device__ __forceinline__ float bce_y0(float x) {
    // BCEWithLogits, label 0:  max(x,0) + log1p(exp(-|x|))
    return fmaxf(x, 0.0f) + log1pf(expf(-fabsf(x)));
}
__device__ __forceinline__ float bce_y1(float x) {
    // BCEWithLogits, label 1:  max(x,0) - x + log1p(exp(-|x|))
    return fmaxf(x, 0.0f) - x + log1pf(expf(-fabsf(x)));
}

__global__ void k_init(float* wsf, float* out) {
    if (blockIdx.x == 0 && threadIdx.x == 0) {
        wsf[0] = 0.0f;
        wsf[1] = 0.0f;
        out[0] = 0.0f;
    }
}

__global__ void k_sum_noise(const float* __restrict__ noise, float* wsf) {
    __shared__ float red[256];
    float s = 0.0f;
    for (int i = blockIdx.x * blockDim.x + threadIdx.x; i < V_DIM;
         i += gridDim.x * blockDim.x)
        s += noise[i];
    red[threadIdx.x] = s;
    __syncthreads();
    for (int o = 128; o > 0; o >>= 1) {
        if (threadIdx.x < o) red[threadIdx.x] += red[threadIdx.x + o];
        __syncthreads();
    }
    if (threadIdx.x == 0) atomicAdd(&wsf[0], red[0]);
}

__global__ void k_sum_probs(const float* __restrict__ noise, float* wsf) {
    __shared__ float red[256];
    const float s1 = wsf[0];          // completed by previous kernel (stream order)
    float s = 0.0f;
    for (int i = blockIdx.x * blockDim.x + threadIdx.x; i < V_DIM;
         i += gridDim.x * blockDim.x)
        s += fmaxf(noise[i] / s1, BACKOFF);
    red[threadIdx.x] = s;
    __syncthreads();
    for (int o = 128; o > 0; o >>= 1) {
        if (threadIdx.x < o) red[threadIdx.x] += red[threadIdx.x + o];
        __syncthreads();
    }
    if (threadIdx.x == 0) atomicAdd(&wsf[1], red[0]);
}

// One block (256 threads) per padded column k: gather emb row, fold constants.
__global__ void k_prep(const float* __restrict__ emb_w,
                       const float* __restrict__ bias_w,
                       const float* __restrict__ noise,
                       const int* __restrict__ noise_idx,
                       float* wsf) {
    const int k = blockIdx.x;                       // 0..511
    float* Brow = wsf + WS_BMAT + (size_t)k * E_DIM;
    if (k < K_DIM) {
        const int idx = noise_idx[k];
        Brow[threadIdx.x] = emb_w[(size_t)idx * E_DIM + threadIdx.x];
        if (threadIdx.x == 0) {
            const float s1 = wsf[0], s2 = wsf[1];
            const float lp = logf(fmaxf(noise[idx] / s1, BACKOFF)) - logf(s2);
            wsf[WS_NCOL + k] = bias_w[idx] - LOG_V - lp - LOG_K;
        }
    } else {
        Brow[threadIdx.x] = 0.0f;                   // zero pad -> harmless WMMA cols
        if (threadIdx.x == 0) wsf[WS_NCOL + k] = 0.0f;
    }
}

// WMMA fp32 GEMM with fused BCE epilogue.
// 32 rows (2 M-subtiles) per workgroup; 8 waves each own a 16x64 column strip
// of both M-subtiles, so every B fragment is reused by 2 WMMAs from registers.
#define MREP  2
#define MROWS (16 * MREP)  // 32
#define LDA   260   // LDS row pitch in floats: 260 % 64 == 4 -> conflict-free b64 reads
__global__ void __launch_bounds__(256)
k_gemm_nce(const float* __restrict__ input,
           const float* __restrict__ wsf,
           float* __restrict__ out) {
    __shared__ float lds_a[MROWS * LDA];
    __shared__ float red[256];

    const float* __restrict__ Bmat = wsf + WS_BMAT;
    const float* __restrict__ ncol = wsf + WS_NCOL;

    const int m0 = blockIdx.x * MROWS;
    const int t  = threadIdx.x;

    // ---- Stage the 32x256 A tile into LDS ----
#if defined(__gfx1250__) && __has_builtin(__builtin_amdgcn_global_load_async_to_lds_b128)
    {
        // 32 rows x 16 chunks(16B) = 512 chunks; 256 threads x 2 chunks each.
        // Destination pitch is LDA floats, so per-chunk LDS addresses handle the pad.
#pragma unroll
        for (int i = 0; i < 2; ++i) {
            const int c   = t + i * 256;
            const int row = c >> 4;
            const int seg = c & 15;
            __builtin_amdgcn_global_load_async_to_lds_b128(
                (__attribute__((address_space(1))) v4i*)
                    (input + (size_t)(m0 + row) * E_DIM + seg * 4),
                (__attribute__((address_space(3))) v4i*)
                    (&lds_a[row * LDA + seg * 4]),
                0, 0);
        }
    }
#if __has_builtin(__builtin_amdgcn_s_wait_asynccnt)
    __builtin_amdgcn_s_wait_asynccnt(0);
#else
    asm volatile("s_wait_asynccnt 0x0" ::: "memory");
#endif
#else
#pragma unroll
    for (int i = 0; i < MROWS; ++i)
        lds_a[i * LDA + t] = input[(size_t)(m0 + i) * E_DIM + t];
#endif
    __syncthreads();

    const int lane  = t & 31;
    const int wv    = t >> 5;
    const int colin = lane & 15;          // A row / B,C column within 16-wide tile
    const int kh    = (lane >> 4) << 1;   // K sub-offset: lanes 0-15 -> 0, 16-31 -> 2

    v8f acc0[4] = {};
    v8f acc1[4] = {};
    const float* __restrict__ bbase = &Bmat[(size_t)(wv * 64 + colin) * E_DIM];

    for (int kk = 0; kk < E_DIM; kk += 4) {
        const int kb = kk + kh;
        const v2f a0 = *(const v2f*)&lds_a[colin * LDA + kb];
        const v2f a1 = *(const v2f*)&lds_a[(16 + colin) * LDA + kb];
#pragma unroll
        for (int q = 0; q < 4; ++q) {
            const v2f b = *(const v2f*)&bbase[(size_t)(q * 16) * E_DIM + kb];
            acc0[q] = __builtin_amdgcn_wmma_f32_16x16x4_f32(
                false, a0, false, b, (short)0, acc0[q], false, false);
            acc1[q] = __builtin_amdgcn_wmma_f32_16x16x4_f32(
                false, a1, false, b, (short)0, acc1[q], false, false);
        }
    }

    // Fused epilogue: logit_true = score + ncol[col]; label 0; sum everything.
    float part = 0.0f;
#pragma unroll
    for (int q = 0; q < 4; ++q) {
        const int col = wv * 64 + q * 16 + colin;
        if (col < K_DIM) {
            const float cc = ncol[col];
#pragma unroll
            for (int r = 0; r < 8; ++r) {
                part += bce_y0(acc0[q][r] + cc);
                part += bce_y0(acc1[q][r] + cc);
            }
        }
    }
    red[t] = part;
    __syncthreads();
    for (int o = 128; o > 0; o >>= 1) {
        if (t < o) red[t] += red[t + o];
        __syncthreads();
    }
    if (t == 0) atomicAdd(out, red[0] * INV_BL);
}

// Target term: one wave per (b,l) row -> gather-dot, wave reduce, y=1 BCE.
__global__ void k_target(const float* __restrict__ input,
                         const float* __restrict__ emb_w,
                         const float* __restrict__ bias_w,
                         const float* __restrict__ noise,
                         const int* __restrict__ target,
                         const float* __restrict__ wsf,
                         float* __restrict__ out) {
    const int lane = threadIdx.x & 31;
    const int gw   = (blockIdx.x * blockDim.x + threadIdx.x) >> 5;
    const int nw   = (gridDim.x * blockDim.x) >> 5;
    const float s1 = wsf[0], s2 = wsf[1];

    for (int row = gw; row < M_TOTAL; row += nw) {
        const int tg = target[row];
        const float* xi = input + (size_t)row * E_DIM;
        const float* we = emb_w + (size_t)tg  * E_DIM;
        float d = 0.0f;
#pragma unroll
        for (int i = 0; i < E_DIM / 32; ++i)
            d += xi[lane + i * 32] * we[lane + i * 32];
#pragma unroll
        for (int o = 16; o > 0; o >>= 1)
            d += __shfl_xor(d, o, 32);
        if (lane == 0) {
            const float lp = logf(fmaxf(noise[tg] / s1, BACKOFF)) - logf(s2);
            const float x  = d + bias_w[tg] - LOG_V - lp - LOG_K;
            atomicAdd(out, bce_y1(x) * INV_BL);
        }
    }
}

extern "C" void kernel_launch(void* const* d_in, const int* in_sizes, int n_in,
                              void* d_out, int out_size, void* d_ws, size_t ws_size,
                              hipStream_t stream) {
    const float* input  = (const float*)d_in[0];   // [B,L,E]
    const float* emb_w  = (const float*)d_in[1];   // [V,E]
    const float* bias_w = (const float*)d_in[2];   // [V]
    const float* noise  = (const float*)d_in[3];   // [V]
    const int*   target = (const int*)d_in[4];     // [B,L]
    const int*   nidx   = (const int*)d_in[5];     // [K]
    float* out = (float*)d_out;
    float* wsf = (float*)d_ws;

    hipLaunchKernelGGL(k_init,      dim3(1),             dim3(1),   0, stream, wsf, out);
    hipLaunchKernelGGL(k_sum_noise, dim3(128),           dim3(256), 0, stream, noise, wsf);
    hipLaunchKernelGGL(k_sum_probs, dim3(128),           dim3(256), 0, stream, noise, wsf);
    hipLaunchKernelGGL(k_prep,      dim3(N_PAD),         dim3(256), 0, stream,
                       emb_w, bias_w, noise, nidx, wsf);
    hipLaunchKernelGGL(k_gemm_nce,  dim3(M_TOTAL/MROWS), dim3(256), 0, stream,
                       input, wsf, out);
    hipLaunchKernelGGL(k_target,    dim3(100),           dim3(256), 0, stream,
                       input, emb_w, bias_w, noise, target, wsf, out);
}